// NeuralRuleEnsemble_31430570672171
// MI455X (gfx1250) — compile-verified
//
#include <hip/hip_runtime.h>
#include <stdint.h>

// Problem constants (from reference)
constexpr int kN = 100000;
constexpr int kD = 256;
constexpr int kR = 128;
constexpr int kP = 4;
constexpr int kK = 8;

constexpr int kTPB    = 128;       // 4 waves, one sample per thread
constexpr int kStride = kD + 1;    // 257 floats: odd stride -> conflict-free LDS gathers

typedef float v2f __attribute__((ext_vector_type(2)));

// ---------------------------------------------------------------------------
// Async-stage one 16B chunk; INST_OFFSET applies to BOTH the LDS address and
// the global address (CDNA5 ISA), so a compile-time offset
// replaces per-chunk VALU address increments entirely.
// ---------------------------------------------------------------------------
template <int C>
__device__ __forceinline__ void stage_row(uint32_t lds_addr, uint32_t g_off,
                                          const float* x) {
  if constexpr (C < kD / 4) {
    asm volatile("global_load_async_to_lds_b128 %0, %1, %2 offset:%c3"
                 :
                 : "v"(lds_addr), "v"(g_off), "s"(x), "i"(C * 16)
                 : "memory");
    stage_row<C + 1>(lds_addr, g_off, x);
  }
}

// ---------------------------------------------------------------------------
// Main kernel (packed-f32 path, Wt = [R][K][P] layout) — defined FIRST so the
// disasm snippet shows its inner loop (verify v_pk_fma_f32 selection).
// ---------------------------------------------------------------------------
__global__ __launch_bounds__(kTPB) void nre_kernel_pk(
    const float* __restrict__ x,    // [N][D]
    const float* __restrict__ Wt,   // [R][K][P] (transposed)
    const float* __restrict__ B,    // [R][P]
    const float* __restrict__ C,    // [R]
    const int*   __restrict__ I,    // [R][K]
    float* __restrict__ out)        // [N]
{
  extern __shared__ float xt[];     // kTPB * kStride floats (131,584 B)

  const int t   = threadIdx.x;
  const int n   = blockIdx.x * kTPB + t;
  const int row = (n < kN) ? n : (kN - 1);   // clamp tail (dup loads harmless)

  // Stage this thread's x row into LDS via async global->LDS DMA.
  {
    const uint32_t lds_addr = (uint32_t)(size_t)(&xt[t * kStride]);
    const uint32_t g_off    = (uint32_t)row * (uint32_t)(kD * sizeof(float));
    stage_row<0>(lds_addr, g_off, x);
  }
  asm volatile("s_wait_asynccnt 0" ::: "memory");

  const float* __restrict__ xrow = &xt[t * kStride];

  float result = 0.0f;

  #pragma unroll 4
  for (int r = 0; r < kR; ++r) {
    const float* wr = Wt + r * (kK * kP);   // [k][p], uniform -> SMEM
    const float* br = B + r * kP;
    const int*   ir = I + r * kK;

    // 8 bank-conflict-free LDS gathers (uniform index + odd row stride)
    float xg[kK];
    #pragma unroll
    for (int k = 0; k < kK; ++k) xg[k] = xrow[ir[k]];

    v2f a01 = {br[0], br[1]};
    v2f a23 = {br[2], br[3]};
    #pragma unroll
    for (int k = 0; k < kK; ++k) {
      v2f xk = {xg[k], xg[k]};                        // splat (opsel)
      v2f w01 = *(const v2f*)(wr + k * kP + 0);       // aligned SGPR pair
      v2f w23 = *(const v2f*)(wr + k * kP + 2);       // aligned SGPR pair
      a01 = __builtin_elementwise_fma(xk, w01, a01);  // v_pk_fma_f32
      a23 = __builtin_elementwise_fma(xk, w23, a23);  // v_pk_fma_f32
    }

    // min_p relu(s_p) == relu(min_p s_p): max(.,0) is monotone
    float m = fminf(fminf(a01.x, a01.y), fminf(a23.x, a23.y));
    m = fmaxf(m, 0.0f);
    result = fmaf(C[r], m, result);
  }

  if (n < kN) out[n] = result;
}

// ---------------------------------------------------------------------------
// Tiny one-shot transpose: W[R][P][K] -> Wt[R][K][P]  (4096 floats, 16 KB).
// Makes the P=4 weights of one k contiguous, so packed-f32 FMA can source
// them as aligned SGPR pairs.
// ---------------------------------------------------------------------------
__global__ void nre_transpose_w(const float* __restrict__ W,
                                float* __restrict__ Wt) {
  int idx = blockIdx.x * blockDim.x + threadIdx.x;   // 0 .. R*P*K-1
  if (idx < kR * kP * kK) {
    int r   = idx / (kP * kK);
    int rem = idx - r * (kP * kK);
    int p   = rem / kK;
    int k   = rem - p * kK;
    Wt[r * (kK * kP) + k * kP + p] = W[idx];
  }
}

// ---------------------------------------------------------------------------
// Fallback (scalar kernel, original W layout) — used only if the workspace is
// too small for the 16 KB transposed weights.
// ---------------------------------------------------------------------------
__global__ __launch_bounds__(kTPB) void nre_kernel_scalar(
    const float* __restrict__ x,
    const float* __restrict__ W,    // [R][P][K]
    const float* __restrict__ B,
    const float* __restrict__ C,
    const int*   __restrict__ I,
    float* __restrict__ out)
{
  extern __shared__ float xt[];

  const int t   = threadIdx.x;
  const int n   = blockIdx.x * kTPB + t;
  const int row = (n < kN) ? n : (kN - 1);

  {
    const uint32_t lds_addr = (uint32_t)(size_t)(&xt[t * kStride]);
    const uint32_t g_off    = (uint32_t)row * (uint32_t)(kD * sizeof(float));
    stage_row<0>(lds_addr, g_off, x);
  }
  asm volatile("s_wait_asynccnt 0" ::: "memory");

  const float* __restrict__ xrow = &xt[t * kStride];
  float result = 0.0f;

  #pragma unroll 2
  for (int r = 0; r < kR; ++r) {
    const float* wr = W + r * (kP * kK);
    const float* br = B + r * kP;
    const int*   ir = I + r * kK;

    float xg[kK];
    #pragma unroll
    for (int k = 0; k < kK; ++k) xg[k] = xrow[ir[k]];

    float a0 = br[0], a1 = br[1], a2 = br[2], a3 = br[3];
    #pragma unroll
    for (int k = 0; k < kK; ++k) {
      a0 = fmaf(xg[k], wr[0 * kK + k], a0);
      a1 = fmaf(xg[k], wr[1 * kK + k], a1);
      a2 = fmaf(xg[k], wr[2 * kK + k], a2);
      a3 = fmaf(xg[k], wr[3 * kK + k], a3);
    }
    float m = fminf(fminf(a0, a1), fminf(a2, a3));
    m = fmaxf(m, 0.0f);
    result = fmaf(C[r], m, result);
  }

  if (n < kN) out[n] = result;
}

extern "C" void kernel_launch(void* const* d_in, const int* in_sizes, int n_in,
                              void* d_out, int out_size, void* d_ws, size_t ws_size,
                              hipStream_t stream) {
  (void)in_sizes; (void)n_in; (void)out_size;

  const float* x = (const float*)d_in[0];
  const float* W = (const float*)d_in[1];
  const float* B = (const float*)d_in[2];
  const float* C = (const float*)d_in[3];
  const int*   I = (const int*)d_in[4];
  float* out = (float*)d_out;

  const int    grid = (kN + kTPB - 1) / kTPB;                   // 782 blocks
  const size_t lds  = (size_t)kTPB * kStride * sizeof(float);   // 131,584 B
  const size_t wt_bytes = (size_t)kR * kP * kK * sizeof(float); // 16 KB

  if (d_ws != nullptr && ws_size >= wt_bytes) {
    float* Wt = (float*)d_ws;
    nre_transpose_w<<<(kR * kP * kK + 255) / 256, 256, 0, stream>>>(W, Wt);
    nre_kernel_pk<<<grid, kTPB, lds, stream>>>(x, Wt, B, C, I, out);
  } else {
    nre_kernel_scalar<<<grid, kTPB, lds, stream>>>(x, W, B, C, I, out);
  }
}